// set_trasformer_63969242906901
// MI455X (gfx1250) — compile-verified
//
#include <hip/hip_runtime.h>
#include <hip/hip_bf16.h>

// ---------------------------------------------------------------------------
// Set2Set graph pooling for MI455X (gfx1250, wave32).
// fp32 throughout; GEMMs use V_WMMA_F32_16X16X4_F32 (full precision, and the
// problem is bandwidth bound: ~1.5 GB total traffic -> ~65 us at 23.3 TB/s,
// so low-precision WMMA would not pay for its accuracy loss).
// Workspace (~216 MB): m[N*H] | e[N] | z[G*4H] | h[G*H] | seg[G+1]
// ---------------------------------------------------------------------------

typedef float v2f __attribute__((ext_vector_type(2)));
typedef float v8f __attribute__((ext_vector_type(8)));

#define FDIM   128
#define HDIM   512
#define GNUM   1024
#define TWO_H  1024
#define FOUR_H 2048

__device__ __forceinline__ float hsig(float x) {
  return fminf(fmaxf(0.2f * x + 0.5f, 0.0f), 1.0f);
}
__device__ __forceinline__ float selu_f(float x) {
  const float scale = 1.0507009873554805f;
  const float alpha = 1.6732632423543772f;
  return x > 0.0f ? scale * x : scale * alpha * (expf(x) - 1.0f);
}

// ---------------------------------------------------------------------------
// C[M x Nc] = A[M x K] @ B[K x Nc] + bias[Nc], all fp32 row-major.
// Block = 128 threads (4 waves). Block tile: 16 rows x 256 cols.
// Wave w owns cols [blockIdx.y*256 + w*64, +64) as 4 WMMA 16x16 tiles.
// V_WMMA_F32_16X16X4_F32; K % 4 == 0, M % 16 == 0, Nc % 256 == 0.
// Frag layouts per ISA 7.12.2:
//   A (16x4): lane L -> row L%16, K pair at 2*(L/16)   -> one float2 load
//   B (4x16): lane L -> col L%16, rows 2*(L/16)(+1)
//   C/D     : vgpr r, lane L -> row r + 8*(L/16), col L%16
// All 8 B scalars of a k-step are loaded into independent registers BEFORE
// the 4 WMMAs so the loadcnt wait drains once per k-step, and B advances by
// incremental pointer bump (no per-iteration 64-bit multiply).
// ---------------------------------------------------------------------------
__global__ void wmma_gemm_bias_f32(const float* __restrict__ A, int lda,
                                   const float* __restrict__ B, int ldb,
                                   const float* __restrict__ bias,
                                   float* __restrict__ C, int ldc, int K) {
  const int lane    = threadIdx.x & 31;
  const int wave    = threadIdx.x >> 5;
  const int l16     = lane & 15;
  const int halfSel = lane >> 4;              // 0 or 1
  const int rowBase = blockIdx.x * 16;
  const int colBase = blockIdx.y * 256 + wave * 64;

  v8f acc[4];
  #pragma unroll
  for (int t = 0; t < 4; ++t) {
    const float bv = bias[colBase + t * 16 + l16];
    #pragma unroll
    for (int r = 0; r < 8; ++r) acc[t][r] = bv;
  }

  const float* aptr = A + (size_t)(rowBase + l16) * lda + 2 * halfSel;
  const float* bptr = B + (size_t)(2 * halfSel) * ldb + colBase + l16;
  const size_t bstep = (size_t)4 * ldb;

  for (int k = 0; k < K; k += 4) {
    const float2 av = *(const float2*)aptr;            // 8B aligned
    float bx[4], by[4];
    #pragma unroll
    for (int t = 0; t < 4; ++t) {                      // issue all 8 loads
      bx[t] = bptr[t * 16];
      by[t] = bptr[t * 16 + ldb];
    }
    v2f a; a.x = av.x; a.y = av.y;
    #pragma unroll
    for (int t = 0; t < 4; ++t) {                      // then 4 back-to-back WMMAs
      v2f b; b.x = bx[t]; b.y = by[t];
      acc[t] = __builtin_amdgcn_wmma_f32_16x16x4_f32(
          /*neg_a=*/false, a, /*neg_b=*/false, b,
          /*c_mod=*/(short)0, acc[t], /*reuse_a=*/false, /*reuse_b=*/false);
    }
    aptr += 4;
    bptr += bstep;
  }

  #pragma unroll
  for (int t = 0; t < 4; ++t) {
    #pragma unroll
    for (int r = 0; r < 8; ++r) {
      const int row = rowBase + r + 8 * halfSel;
      C[(size_t)row * ldc + colBase + t * 16 + l16] = acc[t][r];
    }
  }
}

// seg[g] = first node n with idx[n] >= g  (idx sorted int64). g in [0, G].
__global__ void seg_bounds_kernel(const long long* __restrict__ idx, int n,
                                  int* __restrict__ seg) {
  const int g = blockIdx.x * blockDim.x + threadIdx.x;
  if (g > GNUM) return;
  int lo = 0, hi = n;
  while (lo < hi) {
    const int mid = (lo + hi) >> 1;
    if (idx[mid] < (long long)g) lo = mid + 1; else hi = mid;
  }
  seg[g] = lo;
}

__global__ void zero_f32_kernel(float* __restrict__ p, int n) {
  const int i = blockIdx.x * blockDim.x + threadIdx.x;
  if (i < n) p[i] = 0.0f;
}

// h = o * selu(i * selu(zg));  the f*c0 term is identically zero in the
// reference (c0 is never updated inside the loop).  float4 over j.
__global__ void lstm_gates_kernel(const float* __restrict__ z,
                                  float* __restrict__ h) {
  const int i = blockIdx.x * blockDim.x + threadIdx.x;   // quad index
  if (i >= GNUM * HDIM / 4) return;
  const int g  = i / (HDIM / 4);
  const int j4 = i % (HDIM / 4);
  const float4* zr = (const float4*)(z + (size_t)g * FOUR_H);
  const float4 zi = zr[j4];
  const float4 zg = zr[2 * (HDIM / 4) + j4];
  const float4 zo = zr[3 * (HDIM / 4) + j4];
  float4 hv;
  hv.x = hsig(zo.x) * selu_f(hsig(zi.x) * selu_f(zg.x));
  hv.y = hsig(zo.y) * selu_f(hsig(zi.y) * selu_f(zg.y));
  hv.z = hsig(zo.z) * selu_f(hsig(zi.z) * selu_f(zg.z));
  hv.w = hsig(zo.w) * selu_f(hsig(zi.w) * selu_f(zg.w));
  ((float4*)h)[i] = hv;
}

// e[n] = dot(m[n,:], h[idx[n],:]).  One wave per node; each float4 load
// instruction covers a contiguous 512B span across the wave.
__global__ void node_scores_kernel(const float* __restrict__ m,
                                   const float* __restrict__ h,
                                   const long long* __restrict__ idx, int n,
                                   float* __restrict__ e) {
  const int node = blockIdx.x * 8 + (threadIdx.x >> 5);
  if (node >= n) return;
  const int lane = threadIdx.x & 31;
  const float4* mp = (const float4*)(m + (size_t)node * HDIM);
  const float4* hp = (const float4*)(h + (size_t)idx[node] * HDIM);
  float s = 0.0f;
  #pragma unroll
  for (int j = 0; j < 4; ++j) {
    const float4 a = mp[lane + j * 32];
    const float4 b = hp[lane + j * 32];
    s += a.x * b.x + a.y * b.y + a.z * b.z + a.w * b.w;
  }
  #pragma unroll
  for (int off = 16; off > 0; off >>= 1) s += __shfl_xor(s, off, 32);
  if (lane == 0) e[node] = s;
}

// In-place segment softmax: one 256-thread workgroup per graph.
__global__ void seg_softmax_kernel(float* __restrict__ e,
                                   const int* __restrict__ seg) {
  __shared__ float red[8];
  __shared__ float bcast;
  const int g = blockIdx.x;
  const int s = seg[g], t = seg[g + 1];
  const int lane = threadIdx.x & 31, wv = threadIdx.x >> 5;

  float mx = -3.402823466e38f;
  for (int n = s + (int)threadIdx.x; n < t; n += 256) mx = fmaxf(mx, e[n]);
  #pragma unroll
  for (int off = 16; off > 0; off >>= 1) mx = fmaxf(mx, __shfl_xor(mx, off, 32));
  if (lane == 0) red[wv] = mx;
  __syncthreads();
  if (threadIdx.x == 0) {
    float v = red[0];
    for (int i = 1; i < 8; ++i) v = fmaxf(v, red[i]);
    bcast = v;
  }
  __syncthreads();
  mx = bcast;

  float sm = 0.0f;
  for (int n = s + (int)threadIdx.x; n < t; n += 256) sm += expf(e[n] - mx);
  #pragma unroll
  for (int off = 16; off > 0; off >>= 1) sm += __shfl_xor(sm, off, 32);
  __syncthreads();
  if (lane == 0) red[wv] = sm;
  __syncthreads();
  if (threadIdx.x == 0) {
    float v = 0.0f;
    for (int i = 0; i < 8; ++i) v += red[i];
    bcast = v;
  }
  __syncthreads();
  const float inv = 1.0f / bcast;          // empty graph -> no writes below
  for (int n = s + (int)threadIdx.x; n < t; n += 256)
    e[n] = expf(e[n] - mx) * inv;
}

// r[g,:] = sum_{n in g} a[n]*m[n,:];  q_star = [h | r] -> d_out.
// One 128-thread workgroup per graph; each thread owns 4 consecutive columns,
// so each row of m is read as one contiguous 2KB wavefront-coalesced sweep.
__global__ void seg_weighted_sum_kernel(const float* __restrict__ m,
                                        const float* __restrict__ a,
                                        const float* __restrict__ h,
                                        const int* __restrict__ seg,
                                        float* __restrict__ q) {
  const int g = blockIdx.x;
  const int c4 = threadIdx.x;               // 0..127 -> cols 4*c4..4*c4+3
  const int s = seg[g], t = seg[g + 1];
  float4 acc = make_float4(0.f, 0.f, 0.f, 0.f);
  for (int n = s; n < t; ++n) {
    const float4 mv = ((const float4*)(m + (size_t)n * HDIM))[c4];
    const float w = a[n];
    acc.x += mv.x * w; acc.y += mv.y * w;
    acc.z += mv.z * w; acc.w += mv.w * w;
  }
  float4* qrow = (float4*)(q + (size_t)g * TWO_H);
  qrow[c4]                = ((const float4*)(h + (size_t)g * HDIM))[c4];
  qrow[HDIM / 4 + c4]     = acc;
}

extern "C" void kernel_launch(void* const* d_in, const int* in_sizes, int n_in,
                              void* d_out, int out_size, void* d_ws, size_t ws_size,
                              hipStream_t stream) {
  const float*     features = (const float*)d_in[0];      // (1, N, 128)
  const long long* gidx     = (const long long*)d_in[1];  // (1, N) int64 sorted
  const float*     m_weight = (const float*)d_in[3];      // (128, 512)
  const float*     m_bias   = (const float*)d_in[4];      // (512,)
  const float*     rk       = (const float*)d_in[5];      // (1024, 2048)
  const float*     rbias    = (const float*)d_in[6];      // (2048,)
  float* out = (float*)d_out;                             // (1, 1024, 1024)

  const int N = in_sizes[0] / FDIM;                       // 100000

  // workspace carve-up
  float* m  = (float*)d_ws;                               // N*H
  float* e  = m + (size_t)N * HDIM;                       // N (scores -> attn)
  float* z  = e + (size_t)N;                              // G*4H
  float* h  = z + (size_t)GNUM * FOUR_H;                  // G*H
  int* seg  = (int*)(h + (size_t)GNUM * HDIM);            // G+1

  // 1. segment bounds from sorted graph_index
  seg_bounds_kernel<<<(GNUM + 1 + 255) / 256, 256, 0, stream>>>(gidx, N, seg);

  // 2. m = features @ m_weight + m_bias   (N x 128 @ 128 x 512, fp32 WMMA)
  wmma_gemm_bias_f32<<<dim3(N / 16, HDIM / 256), 128, 0, stream>>>(
      features, FDIM, m_weight, HDIM, m_bias, m, HDIM, FDIM);

  // 3. q_star = 0
  zero_f32_kernel<<<(GNUM * TWO_H + 255) / 256, 256, 0, stream>>>(
      out, GNUM * TWO_H);

  // 4. three Set2Set iterations
  for (int it = 0; it < 3; ++it) {
    // z = q_star @ recurrent_kernel + recurrent_bias (1024x1024 @ 1024x2048)
    wmma_gemm_bias_f32<<<dim3(GNUM / 16, FOUR_H / 256), 128, 0, stream>>>(
        out, TWO_H, rk, FOUR_H, rbias, z, FOUR_H, TWO_H);
    // gates -> h
    lstm_gates_kernel<<<(GNUM * HDIM / 4 + 255) / 256, 256, 0, stream>>>(z, h);
    // e[n] = <m[n], h[idx[n]]>
    node_scores_kernel<<<(N + 7) / 8, 256, 0, stream>>>(m, h, gidx, N, e);
    // segment softmax (in place: e becomes a)
    seg_softmax_kernel<<<GNUM, 256, 0, stream>>>(e, seg);
    // r = segment_sum(a*m); q_star = [h | r]
    seg_weighted_sum_kernel<<<GNUM, 128, 0, stream>>>(m, e, h, seg, out);
  }
}